// RNNHOPF_20856361189389
// MI455X (gfx1250) — compile-verified
//
#include <hip/hip_runtime.h>

// ---------------- problem constants ----------------
#define NN    800
#define CC    64
#define TRS   40
#define SPT   10
#define BUF   120
#define TT    (TRS * SPT)          // 400
#define DTF   1e-4f
#define AF    (-0.5f)
#define OMEGA 10.0f
#define GF    500.0f
#define CY0   5.0f
#define Y0OFF 2.0f
#define KGKI  5.0f                 // KG*KI
#define SQDT_STD 1.0f              // sqrt(1e-4)*100

#define RING  128                  // history ring slots (pow2 > BUF+1)
#define NWG   25                   // workgroups in persistent grid
#define ROWS_PER_WG 32             // 25*32 = 800
#define MAINT 512                  // threads per WG (16 lanes per row x 32 rows)
#define MPAD  48                   // x_tr rows padded to 3 full 16-row M tiles

// ---------------- workspace layout (bytes) ----------------
#define BAR_OFF   0u               // 2 x u32 barrier state
#define PART_OFF  256u             // 256 f32 partial sums
#define INVN_OFF  1536u            // 1 f32
#define RS_OFF    1664u            // 800 f32 row sums
#define H_OFF     5120u            // 800*128 f32 history ring
#define XTR_OFF   417792u          // 48*800 f32 TR samples (rows 40..47 stay zero)
#define PW_OFF    574464u          // 640000 x 8B packed {w,d}

struct __align__(8) WD { float w; int d; };
struct Bar { unsigned count; unsigned gen; };

typedef float v2f __attribute__((ext_vector_type(2)));
typedef float v8f __attribute__((ext_vector_type(8)));

// ---------------- barrier init ----------------
__global__ void k_init_bar(Bar* bar) {
  bar->count = 0u; bar->gen = 0u;
}

// ---------------- Frobenius norm pass 1: per-block sum of sc^2 ----------------
__global__ void k_sumsq(const float* __restrict__ sc, float* __restrict__ partials) {
  float s = 0.f;
  for (int e = blockIdx.x * 256 + threadIdx.x; e < NN * NN; e += 256 * 256) {
    float v = sc[e];
    s = fmaf(v, v, s);
  }
  for (int off = 16; off; off >>= 1) s += __shfl_xor(s, off, 32);
  __shared__ float red[8];
  if ((threadIdx.x & 31) == 0) red[threadIdx.x >> 5] = s;
  __syncthreads();
  if (threadIdx.x == 0) {
    float tot = 0.f;
    for (int k = 0; k < 8; ++k) tot += red[k];
    partials[blockIdx.x] = tot;
  }
}

// ---------------- Frobenius norm pass 2 ----------------
__global__ void k_finish_norm(const float* __restrict__ partials, float* __restrict__ invn) {
  float s = (threadIdx.x < 256) ? partials[threadIdx.x] : 0.f;
  for (int off = 16; off; off >>= 1) s += __shfl_xor(s, off, 32);
  __shared__ float red[8];
  if ((threadIdx.x & 31) == 0) red[threadIdx.x >> 5] = s;
  __syncthreads();
  if (threadIdx.x == 0) {
    float tot = 0.f;
    for (int k = 0; k < 8; ++k) tot += red[k];
    invn[0] = 1.0f / sqrtf(tot);
  }
}

// ---------------- pack w_n + delay into 8-byte records ----------------
__global__ void k_pack(const float* __restrict__ sc, const int* __restrict__ delays,
                       const float* __restrict__ invn, WD* __restrict__ pw) {
  int e = blockIdx.x * 256 + threadIdx.x;
  if (e < NN * NN) {
    WD r;
    r.w = fabsf(sc[e]) * invn[0];
    r.d = delays[e];
    pw[e] = r;
  }
}

// ---------------- row sums of w_n ----------------
__global__ void k_rowsum(const float* __restrict__ sc, const float* __restrict__ invn,
                         float* __restrict__ rowsum) {
  const int l16 = threadIdx.x & 15;
  const int i = blockIdx.x * ROWS_PER_WG + (threadIdx.x >> 4);
  const float inv = invn[0];
  float s = 0.f;
  const float* row = sc + (size_t)i * NN;
  for (int k = 0; k < NN / 16; ++k) s += fabsf(row[k * 16 + l16]) * inv;
  for (int off = 8; off; off >>= 1) s += __shfl_xor(s, off, 16);
  if (l16 == 0) rowsum[i] = s;
}

// ---------------- history ring init: slot (127-d) holds hE[:,d] ----------------
__global__ void k_hist_init(const float* __restrict__ hE, float* __restrict__ H) {
  int idx = blockIdx.x * 256 + threadIdx.x;
  if (idx < NN * RING) {
    int j = idx >> 7, slot = idx & (RING - 1);
    float v = (slot >= RING - BUF) ? hE[j * BUF + (RING - 1 - slot)] : 0.f;
    H[idx] = v;
  }
}

// ---------------- zero-fill padded x_tr staging buffer ----------------
__global__ void k_xtr_zero(float* __restrict__ xtr) {
  int idx = blockIdx.x * 256 + threadIdx.x;
  if (idx < MPAD * NN) xtr[idx] = 0.f;
}

// ---------------- persistent Hopf time-stepper ----------------
__global__ __launch_bounds__(MAINT, 1)
void k_hopf(const float* __restrict__ ext, const float* __restrict__ hx,
            const float* __restrict__ noise, const WD* __restrict__ pw,
            const float* __restrict__ rowsum_g, float* H, float* xtr, Bar* bar) {
  const int tid = threadIdx.x;
  const int rl  = tid >> 4;                  // 0..31 row slot in WG
  const int l16 = tid & 15;
  const int i   = blockIdx.x * ROWS_PER_WG + rl;   // global node owned

  __shared__ float xs_s[ROWS_PER_WG], ys_s[ROWS_PER_WG], rs_s[ROWS_PER_WG];
  if (tid < ROWS_PER_WG) {
    int g = blockIdx.x * ROWS_PER_WG + tid;
    xs_s[tid] = hx[g * 2 + 0];
    ys_s[tid] = hx[g * 2 + 1];
    rs_s[tid] = rowsum_g[g];
  }
  __syncthreads();

  const WD* prow = pw + (size_t)i * NN;

  for (int t = 0; t < TT; ++t) {
    const int q = (RING - 1) + t;            // ring index of newest sample
    float acc = 0.f;
#pragma unroll 5
    for (int k = 0; k < NN / 16; ++k) {
      int j  = k * 16 + l16;
      WD  e  = prow[j];                                        // one b64 load
      int sl = (q - e.d) & (RING - 1);
      // device-scope load: must see other WGPs' writes from the previous step
      float h = __hip_atomic_load(&H[j * RING + sl],
                                  __ATOMIC_RELAXED, __HIP_MEMORY_SCOPE_AGENT);
      acc = fmaf(e.w, h, acc);
    }
    for (int off = 8; off; off >>= 1) acc += __shfl_xor(acc, off, 16);

    if (l16 == 0) {
      float x = xs_s[rl], y = ys_s[rl];
      int s = t % SPT, r = t / SPT;
      float u  = ext[(size_t)i * (SPT * TRS) + s * TRS + r];
      float n0 = noise[(size_t)t * NN * 2 + i * 2 + 0];
      float n1 = noise[(size_t)t * NN * 2 + i * 2 + 1];
      float r2 = x * x + y * y;
      float dx = (AF - r2) * x - OMEGA * y + GF * (acc - rs_s[rl] * x) + KGKI * u;
      float dy = (AF - r2) * y + OMEGA * x;
      float xn = x + DTF * dx + SQDT_STD * n0;
      float yn = y + DTF * dy + SQDT_STD * n1;
      xs_s[rl] = xn; ys_s[rl] = yn;
      // write-slot (q+1) is never read during this step (128 > BUF+1)
      __hip_atomic_store(&H[i * RING + ((q + 1) & (RING - 1))], xn,
                         __ATOMIC_RELAXED, __HIP_MEMORY_SCOPE_AGENT);
      if (s == SPT - 1) xtr[r * NN + i] = xn;
    }

    if (t != TT - 1) {                       // device-wide step barrier
      __threadfence();
      __syncthreads();
      if (tid == 0) {
        unsigned target = (unsigned)(t + 1);
        unsigned arr = __hip_atomic_fetch_add(&bar->count, 1u,
                                              __ATOMIC_ACQ_REL, __HIP_MEMORY_SCOPE_AGENT);
        if (arr == NWG - 1) {
          __hip_atomic_store(&bar->count, 0u, __ATOMIC_RELAXED, __HIP_MEMORY_SCOPE_AGENT);
          __hip_atomic_store(&bar->gen, target, __ATOMIC_RELEASE, __HIP_MEMORY_SCOPE_AGENT);
        } else {
          while (__hip_atomic_load(&bar->gen, __ATOMIC_ACQUIRE,
                                   __HIP_MEMORY_SCOPE_AGENT) < target)
            __builtin_amdgcn_s_sleep(1);
        }
      }
      __syncthreads();
    }
  }
}

// ---------------- leadfield projection via f32 WMMA ----------------
// eeg[c][r] = CY0 * sum_k xtr[r][k]*lm[c][k] - Y0OFF ;  out is (64,40) row-major.
// GEMM: M=48 (padded rows, 3 tiles), N=64 (channels, 4 tiles), K=800 step 4.
// xtr rows 40..47 are zero-padded so all fragment loads are unconditional b64.
__global__ __launch_bounds__(384, 1)
void k_project(const float* __restrict__ xtr, const float* __restrict__ lm,
               float* __restrict__ out) {
  const int w   = threadIdx.x >> 5;          // wave 0..11 -> one 16x16 tile
  const int l   = threadIdx.x & 31;
  const int mt  = w % 3, nt = w / 3;
  const int hi  = l >> 4, n16 = l & 15;
  const int row = mt * 16 + n16;             // A row for this lane (padded, always valid)
  const int col = nt * 16 + n16;             // B column / channel

  const float* ap = xtr + (size_t)row * NN + 2 * hi;   // 8B aligned
  const float* bp = lm  + (size_t)col * NN + 2 * hi;   // 8B aligned

  v8f c = {};
#pragma unroll 4
  for (int k0 = 0; k0 < NN; k0 += 4) {
    // ISA 16x4 f32 A layout: VGPR0 -> K=k0 (lanes 0-15) / k0+2 (lanes 16-31), VGPR1 -> +1
    v2f a = *(const v2f*)(ap + k0);
    v2f b = *(const v2f*)(bp + k0);
    c = __builtin_amdgcn_wmma_f32_16x16x4_f32(false, a, false, b,
                                              (short)0, c, false, false);
  }
#pragma unroll
  for (int v = 0; v < 8; ++v) {
    int m = mt * 16 + v + 8 * hi;            // C/D layout: VGPR v -> M=v (+8 for hi lanes)
    if (m < TRS) out[col * TRS + m] = CY0 * c[v] - Y0OFF;
  }
}

// ---------------- launcher ----------------
extern "C" void kernel_launch(void* const* d_in, const int* in_sizes, int n_in,
                              void* d_out, int out_size, void* d_ws, size_t ws_size,
                              hipStream_t stream) {
  const float* ext   = (const float*)d_in[0];   // (800,10,40)
  const float* hx    = (const float*)d_in[1];   // (800,2)
  const float* hE    = (const float*)d_in[2];   // (800,120)
  const float* sc    = (const float*)d_in[3];   // (800,800)
  const float* lm    = (const float*)d_in[4];   // (64,800)
  const float* noise = (const float*)d_in[5];   // (400,800,2)
  const int*   dly   = (const int*)d_in[6];     // (800,800)

  char* ws = (char*)d_ws;
  Bar*   bar      = (Bar*)(ws + BAR_OFF);
  float* partials = (float*)(ws + PART_OFF);
  float* invn     = (float*)(ws + INVN_OFF);
  float* rowsum   = (float*)(ws + RS_OFF);
  float* H        = (float*)(ws + H_OFF);
  float* xtr      = (float*)(ws + XTR_OFF);
  WD*    pw       = (WD*)(ws + PW_OFF);

  k_init_bar<<<1, 1, 0, stream>>>(bar);
  k_sumsq<<<256, 256, 0, stream>>>(sc, partials);
  k_finish_norm<<<1, 256, 0, stream>>>(partials, invn);
  k_pack<<<(NN * NN + 255) / 256, 256, 0, stream>>>(sc, dly, invn, pw);
  k_rowsum<<<NWG, MAINT, 0, stream>>>(sc, invn, rowsum);
  k_hist_init<<<(NN * RING + 255) / 256, 256, 0, stream>>>(hE, H);
  k_xtr_zero<<<(MPAD * NN + 255) / 256, 256, 0, stream>>>(xtr);
  k_hopf<<<NWG, MAINT, 0, stream>>>(ext, hx, noise, pw, rowsum, H, xtr, bar);
  k_project<<<1, 384, 0, stream>>>(xtr, lm, (float*)d_out);
}